// LSTMLayer_9363028705610
// MI455X (gfx1250) — compile-verified
//
#include <hip/hip_runtime.h>
#include <math.h>

typedef __attribute__((ext_vector_type(2))) float v2f;
typedef __attribute__((ext_vector_type(4))) float v4f;
typedef __attribute__((ext_vector_type(8))) float v8f;

#define S_LEN 512
#define B_DIM 64
#define I_DIM 512
#define H_DIM 512
#define NWG   32                    // 32 WGs x 16 h-columns = H=512
#define WAVES_PER_WG 4              // 4 waves x 16 batch rows = B=64
#define TOTAL_WAVES (NWG * WAVES_PER_WG)

// Gate-interleaved weight cache: [matrix][rsel 0..15][p=K/2 0..255][gate 0..3][2w]
// Row stride padded so 2060 % 64 == 12 -> the 16 lanes' b128 reads tile all
// sixteen 4-bank groups exactly once (upper half-wave offset +8 banks):
// every bank serves exactly 2 lanes = the wave32 b128 bandwidth floor.
#define R_STRIDE 2060                   // 256*8 + 12 pad words
#define M_STRIDE (16 * R_STRIDE)        // per-matrix block (ih | hh)
#define LDS_WORDS (2 * M_STRIDE)        // 65,920 words
#define LDS_BYTES (LDS_WORDS * 4)       // 263,680 B (< 320 KB WGP LDS)

__device__ __forceinline__ float tanh_hw(float x) {
#if __has_builtin(__builtin_amdgcn_tanhf)
    return __builtin_amdgcn_tanhf(x);   // CDNA5 V_TANH_F32 (single trans op)
#else
    return tanhf(x);
#endif
}

__device__ __forceinline__ float sigmoid_hw(float x) {
    // sigmoid(x) = 0.5 + 0.5 * tanh(x/2)  -> 1 trans + 2 VALU
    return __builtin_fmaf(0.5f, tanh_hw(0.5f * x), 0.5f);
}

__device__ __forceinline__ v8f splat8(float v) {
    v8f r;
#pragma unroll
    for (int i = 0; i < 8; ++i) r[i] = v;
    return r;
}

// Zero the per-step barrier slots and the h(-1) zero-state each call so the
// persistent kernel is deterministic across graph replays.
__global__ void lstm_init_kernel(unsigned* __restrict__ slots,
                                 float* __restrict__ h0) {
    int i = blockIdx.x * blockDim.x + threadIdx.x;
    if (i < S_LEN) slots[i] = 0u;
    const int n = B_DIM * H_DIM;
    for (int j = i; j < n; j += gridDim.x * blockDim.x) {
        h0[j] = 0.0f;
    }
}

__global__ void __launch_bounds__(128, 1)
lstm_persistent_kernel(const float* __restrict__ x,     // [S,B,I]
                       const float* __restrict__ w_ih,  // [4H,I]
                       const float* __restrict__ w_hh,  // [4H,H]
                       const float* __restrict__ b_ih,  // [4H]
                       const float* __restrict__ b_hh,  // [4H]
                       float* __restrict__ out,         // [S,B,H] ++ h[B,H] ++ c[B,H]
                       unsigned* __restrict__ slots,    // [S]
                       const float* __restrict__ h0)    // [B,H] zeros (h at t=-1)
{
    extern __shared__ float lds_w[];

    const int lane    = threadIdx.x & 31;
    const int wave    = threadIdx.x >> 5;   // 0..3 : batch row block
    const int rowbase = wave * 16;
    const int c0      = blockIdx.x * 16;    // owned h-column block
    const int rsel    = lane & 15;          // row/col selector within fragment
    const int ksel    = (lane >> 4) * 2;    // K sub-offset for A/B fragments
    const int rowoff  = (lane >> 4) * 8;    // C/D tile: upper half-wave -> M+8

    // ---- stage weight slice into gate-interleaved LDS (once) --------------
    {
        const int tid = threadIdx.x;        // one 512-float weight row per thread
        const int m   = tid >> 6;           // 0 = w_ih, 1 = w_hh
        const int t   = (tid >> 4) & 3;     // gate group
        const int n   = tid & 15;           // column within group
        const float* src = (m == 0 ? w_ih : w_hh)
                           + (size_t)(t * H_DIM + c0 + n) * I_DIM;  // I==H==512
        float* dst = lds_w + m * M_STRIDE + n * R_STRIDE + t * 2;
        for (int kk = 0; kk < I_DIM; kk += 2) {
            v2f v = *reinterpret_cast<const v2f*>(src + kk);
            *reinterpret_cast<v2f*>(dst + (kk >> 1) * 8) = v;  // ds_store_b64
        }
    }
    __syncthreads();

    // Per-lane LDS base (words); all k offsets become DS immediates.
    const int lbih = rsel * R_STRIDE + (lane >> 4) * 8;
    const int lbhh = M_STRIDE + lbih;

    float bsum[4];
#pragma unroll
    for (int t = 0; t < 4; ++t) {
        const int g = t * H_DIM + c0;
        bsum[t] = b_ih[g + rsel] + b_hh[g + rsel];  // bias for this lane's N-col
    }

    v8f c_state = {};   // cell state tile lives in VGPRs for all 512 steps
    v8f h_tile  = {};

    for (int s = 0; s < S_LEN; ++s) {
        // h(s-1) lives in the output tensor itself; step 0 reads the zero buffer.
        const float* hsrc = (s == 0) ? h0
                                     : out + (size_t)(s - 1) * (B_DIM * H_DIM);

        // Accumulators pre-seeded with bias (per-lane-constant column bias).
        v8f acc0 = splat8(bsum[0]);
        v8f acc1 = splat8(bsum[1]);
        v8f acc2 = splat8(bsum[2]);
        v8f acc3 = splat8(bsum[3]);

        // ---- input projection slice: gates += x[s] @ w_ih^T -------------
        const float* ax = x + ((size_t)s * B_DIM + rowbase + rsel) * I_DIM + ksel;
        if (s + 1 < S_LEN) {
            __builtin_prefetch(ax + (size_t)B_DIM * I_DIM, 0, 3);  // next step's rows
        }
        for (int k = 0; k < I_DIM; k += 4) {
            v2f a  = *reinterpret_cast<const v2f*>(ax + k);
            v4f q0 = *reinterpret_cast<const v4f*>(&lds_w[lbih + (k >> 1) * 8]);     // gates 0,1
            v4f q1 = *reinterpret_cast<const v4f*>(&lds_w[lbih + (k >> 1) * 8 + 4]); // gates 2,3
            v2f b0 = __builtin_shufflevector(q0, q0, 0, 1);
            v2f b1 = __builtin_shufflevector(q0, q0, 2, 3);
            v2f b2 = __builtin_shufflevector(q1, q1, 0, 1);
            v2f b3 = __builtin_shufflevector(q1, q1, 2, 3);
            acc0 = __builtin_amdgcn_wmma_f32_16x16x4_f32(false, a, false, b0, (short)0, acc0, false, false);
            acc1 = __builtin_amdgcn_wmma_f32_16x16x4_f32(false, a, false, b1, (short)0, acc1, false, false);
            acc2 = __builtin_amdgcn_wmma_f32_16x16x4_f32(false, a, false, b2, (short)0, acc2, false, false);
            acc3 = __builtin_amdgcn_wmma_f32_16x16x4_f32(false, a, false, b3, (short)0, acc3, false, false);
        }

        // ---- recurrent projection slice: gates += h(s-1) @ w_hh^T -------
        const float* ah = hsrc + (size_t)(rowbase + rsel) * H_DIM + ksel;
        for (int k = 0; k < H_DIM; k += 4) {
            v2f a  = *reinterpret_cast<const v2f*>(ah + k);
            v4f q0 = *reinterpret_cast<const v4f*>(&lds_w[lbhh + (k >> 1) * 8]);
            v4f q1 = *reinterpret_cast<const v4f*>(&lds_w[lbhh + (k >> 1) * 8 + 4]);
            v2f b0 = __builtin_shufflevector(q0, q0, 0, 1);
            v2f b1 = __builtin_shufflevector(q0, q0, 2, 3);
            v2f b2 = __builtin_shufflevector(q1, q1, 0, 1);
            v2f b3 = __builtin_shufflevector(q1, q1, 2, 3);
            acc0 = __builtin_amdgcn_wmma_f32_16x16x4_f32(false, a, false, b0, (short)0, acc0, false, false);
            acc1 = __builtin_amdgcn_wmma_f32_16x16x4_f32(false, a, false, b1, (short)0, acc1, false, false);
            acc2 = __builtin_amdgcn_wmma_f32_16x16x4_f32(false, a, false, b2, (short)0, acc2, false, false);
            acc3 = __builtin_amdgcn_wmma_f32_16x16x4_f32(false, a, false, b3, (short)0, acc3, false, false);
        }

        // ---- elementwise LSTM cell update (hardware tanh, registers) ----
#pragma unroll
        for (int r = 0; r < 8; ++r) {
            float ig = sigmoid_hw(acc0[r]);
            float fg = sigmoid_hw(acc1[r]);
            float gg = tanh_hw(acc2[r]);
            float og = sigmoid_hw(acc3[r]);
            float cc = __builtin_fmaf(fg, c_state[r], ig * gg);
            c_state[r] = cc;
            h_tile[r]  = og * tanh_hw(cc);
        }

        // ---- scatter h tile to outputs[s] (doubles as h history) --------
        {
            float* outp = out + (size_t)s * (B_DIM * H_DIM);
            const int col = c0 + rsel;
#pragma unroll
            for (int r = 0; r < 8; ++r) {
                outp[(size_t)(rowbase + rowoff + r) * H_DIM + col] = h_tile[r];
            }
        }

        // ---- grid-wide barrier: per-step slot, agent scope --------------
        if (lane == 0) {
            __hip_atomic_fetch_add(&slots[s], 1u, __ATOMIC_ACQ_REL,
                                   __HIP_MEMORY_SCOPE_AGENT);
        }
        while (__hip_atomic_load(&slots[s], __ATOMIC_ACQUIRE,
                                 __HIP_MEMORY_SCOPE_AGENT) < (unsigned)TOTAL_WAVES) {
            __builtin_amdgcn_s_sleep(1);
        }
    }

    // ---- final h and c tiles ---------------------------------------------
    float* hout = out + (size_t)S_LEN * B_DIM * H_DIM;
    float* cout = hout + B_DIM * H_DIM;
    const int col = c0 + rsel;
#pragma unroll
    for (int r = 0; r < 8; ++r) {
        const int row = rowbase + rowoff + r;
        hout[(size_t)row * H_DIM + col] = h_tile[r];
        cout[(size_t)row * H_DIM + col] = c_state[r];
    }
}

extern "C" void kernel_launch(void* const* d_in, const int* in_sizes, int n_in,
                              void* d_out, int out_size, void* d_ws, size_t ws_size,
                              hipStream_t stream) {
    const float* x    = (const float*)d_in[0];
    const float* wih  = (const float*)d_in[1];
    const float* whh  = (const float*)d_in[2];
    const float* bih  = (const float*)d_in[3];
    const float* bhh  = (const float*)d_in[4];
    float* out = (float*)d_out;

    unsigned* slots = (unsigned*)d_ws;
    float* h0 = (float*)d_ws + S_LEN;          // 2KB in, 8B-aligned

    // Allow >64KB dynamic LDS (CDNA5 WGP has 320KB). Idempotent, capture-safe.
    (void)hipFuncSetAttribute((const void*)lstm_persistent_kernel,
                              hipFuncAttributeMaxDynamicSharedMemorySize,
                              LDS_BYTES);

    lstm_init_kernel<<<64, 256, 0, stream>>>(slots, h0);
    lstm_persistent_kernel<<<NWG, 128, LDS_BYTES, stream>>>(x, wih, whh, bih, bhh,
                                                            out, slots, h0);
}